// MatrixVT_15719580304175
// MI455X (gfx1250) — compile-verified
//
#include <hip/hip_runtime.h>
#include <hip/hip_bf16.h>
#include <cstdint>

typedef __attribute__((ext_vector_type(2))) float v2f;
typedef __attribute__((ext_vector_type(4))) float v4f;
typedef __attribute__((ext_vector_type(8))) float v8f;
typedef __attribute__((ext_vector_type(4))) int v4i;

#define BB 4
#define NN 6
#define CC 80
#define DD 112
#define HH 16
#define WW 44
#define HW 704      // H*W
#define NK 264      // N*W
#define KP 272      // NK padded to 17*16
#define NKF 17      // KP/16 k-fragments
#define NCH 28      // DD/4 K-chunks
#define XY 16384
#define VT 64       // BEV v-tile per workgroup

#if defined(__has_builtin)
#if __has_builtin(__builtin_amdgcn_global_load_async_to_lds_b128) && \
    __has_builtin(__builtin_amdgcn_s_wait_asynccnt)
#define HAVE_ASYNC_LDS 1
#endif
#endif

__device__ __forceinline__ void copy16_g2l(const float* g, float* l) {
#if defined(HAVE_ASYNC_LDS)
  __builtin_amdgcn_global_load_async_to_lds_b128(
      (__attribute__((address_space(1))) v4i*)(float*)g,
      (__attribute__((address_space(3))) v4i*)l, 0, 0);
#else
  *(float4*)l = *(const float4*)g;
#endif
}

__device__ __forceinline__ void async_join() {
#if defined(HAVE_ASYNC_LDS)
  __builtin_amdgcn_s_wait_asynccnt(0);
#endif
}

// ---------------------------------------------------------------------------
// conv1: feature[bn,80,16,44] -> relu(conv3x3) -> h[bn,64,16,44]
// ---------------------------------------------------------------------------
__global__ void __launch_bounds__(256) conv1_kernel(const float* __restrict__ feat,
                                                    const float* __restrict__ w1,
                                                    const float* __restrict__ b1,
                                                    float* __restrict__ hout) {
  extern __shared__ float smem[];
  float* f_s = smem;           // 80*704 = 56320 floats
  float* w_s = smem + CC * HW; // 720 floats
  int bn = blockIdx.x;  // 0..23
  int oc = blockIdx.y;  // 0..63
  const float* fb = feat + (size_t)bn * CC * HW;
  for (int i = threadIdx.x; i < CC * HW / 4; i += blockDim.x)
    copy16_g2l(fb + 4 * i, f_s + 4 * i);
  const float* wb = w1 + (size_t)oc * CC * 9;
  for (int i = threadIdx.x; i < CC * 9; i += blockDim.x) w_s[i] = wb[i];
  async_join();
  __syncthreads();
  float bias = b1[oc];
  for (int p = threadIdx.x; p < HW; p += blockDim.x) {
    int y = p / WW, x = p % WW;
    float acc = bias;
    for (int ic = 0; ic < CC; ++ic) {
      const float* fc = f_s + ic * HW;
      const float* wc = w_s + ic * 9;
#pragma unroll
      for (int ky = 0; ky < 3; ++ky) {
        int yy = y + ky - 1;
        if (yy < 0 || yy >= HH) continue;
#pragma unroll
        for (int kx = 0; kx < 3; ++kx) {
          int xx = x + kx - 1;
          if (xx < 0 || xx >= WW) continue;
          acc = fmaf(fc[yy * WW + xx], wc[ky * 3 + kx], acc);
        }
      }
    }
    hout[((size_t)bn * 64 + oc) * HW + p] = fmaxf(acc, 0.f);
  }
}

// ---------------------------------------------------------------------------
// conv2: h[bn,64,16,44] -> vlog[bn,16,44]
// ---------------------------------------------------------------------------
__global__ void conv2_kernel(const float* __restrict__ hmid,
                             const float* __restrict__ w2,
                             const float* __restrict__ b2,
                             float* __restrict__ vout) {
  int idx = blockIdx.x * blockDim.x + threadIdx.x;
  if (idx >= 24 * HW) return;
  int x = idx % WW;
  int y = (idx / WW) % HH;
  int bn = idx / HW;
  float acc = b2[0];
  const float* hb = hmid + (size_t)bn * 64 * HW;
  for (int ic = 0; ic < 64; ++ic) {
    const float* hc = hb + ic * HW;
    const float* wc = w2 + ic * 9;
#pragma unroll
    for (int ky = 0; ky < 3; ++ky) {
      int yy = y + ky - 1;
      if (yy < 0 || yy >= HH) continue;
#pragma unroll
      for (int kx = 0; kx < 3; ++kx) {
        int xx = x + kx - 1;
        if (xx < 0 || xx >= WW) continue;
        acc = fmaf(hc[yy * WW + xx], wc[ky * 3 + kx], acc);
      }
    }
  }
  vout[idx] = acc;  // [bn][y][x]
}

// ---------------------------------------------------------------------------
// zero the (padded, swizzled) depth_p workspace: [B][17][32 lanes][56]
// ---------------------------------------------------------------------------
__global__ void zero_kernel(float* __restrict__ dpws) {
  int i = blockIdx.x * blockDim.x + threadIdx.x;
  const int total = BB * NKF * 32 * (2 * NCH);  // 121856
  if (i < total) dpws[i] = 0.f;
}

// ---------------------------------------------------------------------------
// reduce: per (bn,w) column: softmaxes + H reductions.
// depth_p is written PRE-SWIZZLED into the WMMA 16x4-f32 A-fragment layout:
//   k row -> kf = k>>4, r = k&15 ; element d = 4j + 2ku + e
//   addr = (((b*17 + kf)*32) + r + 16*ku)*56 + 2*j + e
// so each (fragment, lane) reads 56 contiguous floats (14 x b128).
// ---------------------------------------------------------------------------
__global__ void __launch_bounds__(128) reduce_kernel(const float* __restrict__ dlog,
                                                     const float* __restrict__ feat,
                                                     const float* __restrict__ vlog,
                                                     float* __restrict__ dp_ws,
                                                     float* __restrict__ fp_ws) {
  __shared__ float s_w[HH];
  __shared__ float s_red[128];
  int bn = blockIdx.x / WW;
  int w = blockIdx.x % WW;
  int tid = threadIdx.x;
  int b = bn / NN, n = bn % NN;
  int k = n * WW + w;

  float vl = (tid < HH) ? vlog[(size_t)bn * HW + tid * WW + w] : 0.f;
  if (tid < HH) s_red[tid] = vl;
  __syncthreads();
  float m = -1e30f;
  for (int i = 0; i < HH; ++i) m = fmaxf(m, s_red[i]);
  float ssum = 0.f;
  for (int i = 0; i < HH; ++i) ssum += __expf(s_red[i] - m);
  if (tid < HH) s_w[tid] = __expf(vl - m) / ssum;
  __syncthreads();

  int d = tid;
  float accd = 0.f;
  for (int h = 0; h < HH; ++h) {
    float lg = (d < DD) ? dlog[(((size_t)bn * DD + d) * HH + h) * WW + w] : -1e30f;
    s_red[tid] = lg;
    __syncthreads();
    for (int s = 64; s > 0; s >>= 1) {
      if (tid < s) s_red[tid] = fmaxf(s_red[tid], s_red[tid + s]);
      __syncthreads();
    }
    float mx = s_red[0];
    __syncthreads();
    float e = (d < DD) ? __expf(lg - mx) : 0.f;
    s_red[tid] = e;
    __syncthreads();
    for (int s = 64; s > 0; s >>= 1) {
      if (tid < s) s_red[tid] += s_red[tid + s];
      __syncthreads();
    }
    float denom = s_red[0];
    __syncthreads();
    accd += (e / denom) * s_w[h];
  }
  if (d < DD) {
    int kf = k >> 4, r = k & 15;
    int j = d >> 2, ku = (d >> 1) & 1, e = d & 1;
    dp_ws[((((size_t)b * NKF + kf) * 32) + r + 16 * ku) * (2 * NCH) + 2 * j + e] = accd;
  }

  if (tid < CC) {
    float mm = -1e30f;
    for (int h = 0; h < HH; ++h)
      mm = fmaxf(mm, feat[(((size_t)bn * CC + tid) * HH + h) * WW + w]);
    fp_ws[((size_t)b * NK + k) * CC + tid] = mm;
  }
}

// ---------------------------------------------------------------------------
// bev: fused double-GEMM with f32 WMMA.
//  Stage A: proj_s[272,64] = (depth_p[272x112] @ circle[112x64]) * ray[272x64]
//  Stage B: out[64x80]     = proj_sT[64x272] @ feat_s[272x80]
// grid = (XY/VT, B); block = 256 (8 wave32 waves); dynamic LDS ~181 KB.
// ---------------------------------------------------------------------------
__global__ void __launch_bounds__(256) bev_kernel(const float* __restrict__ circle,
                                                  const float* __restrict__ ray,
                                                  const float* __restrict__ dp,
                                                  const float* __restrict__ fp,
                                                  float* __restrict__ out) {
  extern __shared__ float smem[];
  float* circ_s = smem;                              // DD*VT  = 7168
  float* proj_s = smem + DD * VT;                    // KP*VT  = 17408
  float* feat_s = smem + DD * VT + (size_t)KP * VT;  // KP*CC  = 21760
  int b = blockIdx.y;
  int v_base = blockIdx.x * VT;
  int tid = threadIdx.x;
  int wave = tid >> 5;
  int lane = tid & 31;
  int lh = lane & 15;   // row/col within 16
  int ku = lane >> 4;   // K-half select (0/1)

  // stage circle tile [112 x 64] into LDS (async where available)
  const float* cb = circle + (size_t)b * DD * XY + v_base;
  for (int i = tid; i < DD * VT / 4; i += 256) {
    int dd = i >> 4;  // VT/4 == 16 quads per row
    int vq = i & 15;
    copy16_g2l(cb + (size_t)dd * XY + vq * 4, circ_s + dd * VT + vq * 4);
  }
  // stage feat_p tile [272 x 80] into LDS (zero padded K rows)
  const float* fb = fp + (size_t)b * NK * CC;
  for (int i = tid; i < KP * CC / 4; i += 256) {
    int kk = i / (CC / 4);
    int cq = i % (CC / 4);
    if (kk < NK) {
      copy16_g2l(fb + (size_t)kk * CC + cq * 4, feat_s + kk * CC + cq * 4);
    } else {
      *(float4*)(feat_s + kk * CC + cq * 4) = make_float4(0.f, 0.f, 0.f, 0.f);
    }
  }
  // warm L2 for the ray tile we're about to stream
  const float* rb = ray + (size_t)b * NK * XY + v_base;
  for (int i = tid; i < NK; i += 256) __builtin_prefetch(rb + (size_t)i * XY, 0, 0);
  async_join();
  __syncthreads();

  // ---- Stage A ----
  // wave w owns vf = w&3 and kf = (w>>2), (w>>2)+2, ... < 17.
  // Circle B-fragments depend only on (d-chunk, vf): hoist all 28 into regs.
  {
    int vf = wave & 3;
    v2f bc[NCH];
#pragma unroll
    for (int j = 0; j < NCH; ++j) {
      int co = (4 * j + 2 * ku) * VT + vf * 16 + lh;
      bc[j].x = circ_s[co];
      bc[j].y = circ_s[co + VT];
    }
    for (int kf = (wave >> 2); kf < NKF; kf += 2) {
      // pre-swizzled A fragment: this lane's 56 floats are contiguous
      const float* dpr = dp + ((((size_t)b * NKF + kf) * 32) + lane) * (2 * NCH);
      v8f acc = {};
#pragma unroll
      for (int jj = 0; jj < NCH / 2; ++jj) {
        v4f q = *(const v4f*)(dpr + 4 * jj);  // chunks 2jj and 2jj+1
        v2f a0 = {q.x, q.y};
        v2f a1 = {q.z, q.w};
        acc = __builtin_amdgcn_wmma_f32_16x16x4_f32(false, a0, false, bc[2 * jj],
                                                    (short)0, acc, false, false);
        acc = __builtin_amdgcn_wmma_f32_16x16x4_f32(false, a1, false, bc[2 * jj + 1],
                                                    (short)0, acc, false, false);
      }
#pragma unroll
      for (int r = 0; r < 8; ++r) {
        int krow = kf * 16 + r + 8 * ku;
        int kr = krow < NK ? krow : (NK - 1);  // clamp: pad rows have acc==0 anyway
        float rv = rb[(size_t)kr * XY + vf * 16 + lh];
        proj_s[krow * VT + vf * 16 + lh] = acc[r] * rv;
      }
    }
  }
  __syncthreads();

  // ---- Stage B: 4x5 fragments of 16x16, K=272 (all-LDS operands) ----
  for (int f = wave; f < (VT / 16) * (CC / 16); f += 8) {
    int vf = f / 5;  // 0..3
    int cf = f % 5;  // 0..4
    v8f acc = {};
    for (int k0 = 0; k0 < KP; k0 += 4) {
      int kk = k0 + 2 * ku;
      v2f a;
      a.x = proj_s[kk * VT + vf * 16 + lh];
      a.y = proj_s[(kk + 1) * VT + vf * 16 + lh];
      v2f bb;
      bb.x = feat_s[kk * CC + cf * 16 + lh];
      bb.y = feat_s[(kk + 1) * CC + cf * 16 + lh];
      acc = __builtin_amdgcn_wmma_f32_16x16x4_f32(false, a, false, bb, (short)0, acc,
                                                  false, false);
    }
    int c = cf * 16 + lh;
    float* ob = out + ((size_t)b * CC + c) * XY + v_base + vf * 16 + 8 * ku;
    // 8 consecutive v per lane -> two b128 stores
    *(float4*)(ob) = make_float4(acc[0], acc[1], acc[2], acc[3]);
    *(float4*)(ob + 4) = make_float4(acc[4], acc[5], acc[6], acc[7]);
  }
}

// ---------------------------------------------------------------------------
extern "C" void kernel_launch(void* const* d_in, const int* in_sizes, int n_in,
                              void* d_out, int out_size, void* d_ws, size_t ws_size,
                              hipStream_t stream) {
  const float* feature = (const float*)d_in[0];  // [24,80,16,44]
  const float* dlog    = (const float*)d_in[1];  // [24,112,16,44]
  const float* circle  = (const float*)d_in[2];  // [4,112,16384]
  const float* ray     = (const float*)d_in[3];  // [4,264,16384]
  const float* w1      = (const float*)d_in[4];  // [64,80,3,3]
  const float* b1      = (const float*)d_in[5];  // [64]
  const float* w2      = (const float*)d_in[6];  // [1,64,3,3]
  const float* b2      = (const float*)d_in[7];  // [1]
  float* out = (float*)d_out;                    // [4,80,128,128]

  float* ws   = (float*)d_ws;
  float* h_ws = ws;                                    // 24*64*704 = 1081344
  float* v_ws = h_ws + (size_t)24 * 64 * HW;           // 24*704    = 16896
  float* dpws = v_ws + (size_t)24 * HW;                // 4*17*32*56 = 121856 (swizzled)
  float* fpws = dpws + (size_t)BB * NKF * 32 * (2 * NCH);  // 4*264*80 = 84480

  {
    dim3 grid(24, 64);
    size_t sm = (size_t)(CC * HW + CC * 9) * sizeof(float);  // ~228 KB
    conv1_kernel<<<grid, 256, sm, stream>>>(feature, w1, b1, h_ws);
  }
  {
    int total = 24 * HW;
    conv2_kernel<<<(total + 255) / 256, 256, 0, stream>>>(h_ws, w2, b2, v_ws);
  }
  zero_kernel<<<(BB * NKF * 32 * (2 * NCH) + 255) / 256, 256, 0, stream>>>(dpws);
  reduce_kernel<<<24 * WW, 128, 0, stream>>>(dlog, feature, v_ws, dpws, fpws);
  {
    dim3 grid(XY / VT, BB);
    size_t sm = (size_t)(DD * VT + KP * VT + KP * CC) * sizeof(float);  // ~181 KB
    bev_kernel<<<grid, 256, sm, stream>>>(circle, ray, dpws, fpws, out);
  }
}